// TRFAligner_27135603376403
// MI455X (gfx1250) — compile-verified
//
#include <hip/hip_runtime.h>

// Problem constants (fixed by the reference).
#define B_DIM   4
#define C_DIM   64
#define W_DIM   64
#define S_DIM   4096
#define NREAL   32768
#define T_TILE  256          // output samples per tile; divides NREAL exactly
#define NTHREADS 256

typedef __attribute__((ext_vector_type(4))) unsigned int v4u;
typedef __attribute__((ext_vector_type(8))) int          v8i;
typedef __attribute__((ext_vector_type(4))) int          v4i;

// One workgroup owns out[b, 0:64, t0:t0+256].
// LDS: 64x256 f32 accumulator (64KB) + staged local positions (16KB worst case).
__global__ __launch_bounds__(NTHREADS) void trf_scatter_kernel(
    const float* __restrict__ trf,          // (B, C, W, S) f32
    const long long* __restrict__ srcIdx,   // (B, S) i64, sorted per batch
    float* __restrict__ out)                // (B, C, NREAL) f32
{
    __shared__ float acc[C_DIM * T_TILE];   // row-major: acc[c*T_TILE + p]
    __shared__ int   dloc[S_DIM];           // idx[s] - t0 for s in [sLo, sHi)
    __shared__ int   sRange[2];

    const int b   = blockIdx.y;
    const int t0  = blockIdx.x * T_TILE;
    const int tid = threadIdx.x;

    // Zero the accumulator tile.
    for (int i = tid; i < C_DIM * T_TILE; i += NTHREADS) acc[i] = 0.0f;

    // Binary-search the contiguous source range touching this tile:
    // sources with idx in [t0-63, t0+T_TILE).
    if (tid == 0) {
        const long long* id = srcIdx + (size_t)b * S_DIM;
        const long long loV = (long long)(t0 - (W_DIM - 1));
        const long long hiV = (long long)(t0 + T_TILE);
        int lo = 0, n = S_DIM;
        while (n > 0) { int h = n >> 1; int m = lo + h;
            if (id[m] < loV) { lo = m + 1; n -= h + 1; } else n = h; }
        int hi = lo; n = S_DIM - lo;
        while (n > 0) { int h = n >> 1; int m = hi + h;
            if (id[m] < hiV) { hi = m + 1; n -= h + 1; } else n = h; }
        sRange[0] = lo; sRange[1] = hi;
    }
    __syncthreads();
    const int sLo = sRange[0];
    const int nS  = sRange[1] - sRange[0];

    // Stage tile-local positions into LDS (reused 4096x by the hot loop).
    for (int j = tid; j < nS; j += NTHREADS)
        dloc[j] = (int)(srcIdx[(size_t)b * S_DIM + sLo + j]) - t0;  // [-63, T)
    __syncthreads();

    // Hot loop: waves stride over (c,w); lanes stride over contiguous s
    // => fully coalesced TRF reads; ds_add_f32 accumulates into the LDS tile.
    const int wave = tid >> 5;
    const int lane = tid & 31;
    for (int cw = wave; cw < C_DIM * W_DIM; cw += (NTHREADS / 32)) {
        const int c = cw >> 6;
        const int w = cw & (W_DIM - 1);
        const float* src =
            trf + ((((size_t)b * C_DIM + c) * W_DIM + w) * S_DIM + sLo);
        for (int j = lane; j < nS; j += 32) {
            const int p = dloc[j] + w;                 // position in tile
            if ((unsigned)p < (unsigned)T_TILE)
                atomicAdd(&acc[c * T_TILE + p], src[j]);   // -> ds_add_f32
        }
    }
    __syncthreads();

    // Write-out via the Tensor Data Mover: one wave issues a 2D
    // tensor_store_from_lds of the 64x256 tile (row stride NREAL floats).
    if (tid < 32) {
        const unsigned lds_off =
            (unsigned)(unsigned long long)(const void*)&acc[0]; // low 32b = LDS offset
        const unsigned long long ga =
            (unsigned long long)(out + ((size_t)b * C_DIM) * NREAL + t0);

        // D# group 0: count=1 | lds_addr | global_addr(57b) | type=2
        v4u g0 = { 1u,
                   lds_off,
                   (unsigned)(ga & 0xFFFFFFFFu),
                   (unsigned)((ga >> 32) & 0x01FFFFFFu) | (2u << 30) };
        // D# group 1:
        //  s0: wg_mask=0, data_size=2 (4B)
        //  s1[31:16]=tensor_dim0.lo16=256   s2[31:16]=tensor_dim1.lo16=64
        //  s3[31:16]=tile_dim0=256          s4[15:0]=tile_dim1=64
        //  s5=tensor_dim0_stride=32768 (floats)
        v8i g1 = { (int)0x00020000,
                   (int)(256u << 16),
                   (int)(64u  << 16),
                   (int)(256u << 16),
                   64,
                   32768,
                   0, 0 };
        v4i g2 = { 0, 0, 0, 0 };   // 2D tensor: groups 2/3 unused
        v4i g3 = { 0, 0, 0, 0 };
        v8i g4 = { 0, 0, 0, 0, 0, 0, 0, 0 };   // extra group (clang-23 form)
        __builtin_amdgcn_tensor_store_from_lds(g0, g1, g2, g3, g4, 0);
        __builtin_amdgcn_s_wait_tensorcnt(0);
    }
}

extern "C" void kernel_launch(void* const* d_in, const int* in_sizes, int n_in,
                              void* d_out, int out_size, void* d_ws, size_t ws_size,
                              hipStream_t stream) {
    const float*     trf = (const float*)d_in[0];      // (4,64,64,4096) f32
    const long long* idx = (const long long*)d_in[1];  // (4,4096) i64 (sorted)
    float*           out = (float*)d_out;              // (4,64,32768) f32

    dim3 grid(NREAL / T_TILE, B_DIM, 1);               // 128 x 4 = 512 workgroups
    trf_scatter_kernel<<<grid, NTHREADS, 0, stream>>>(trf, idx, out);
}